// CostTokenizer_89189290869215
// MI455X (gfx1250) — compile-verified
//
#include <hip/hip_runtime.h>

typedef __attribute__((ext_vector_type(2))) float v2f;
typedef __attribute__((ext_vector_type(8))) float v8f;

// One block = 128 threads (4 waves, wave32). Each block produces a 64-pixel
// tile of the [T-1,B,192,H,W] output for one (t,b) pair.
//   Phase 1: corr[64 pixels][49 offsets] into LDS (VALU dots over C).
//   Phase 2: tok^T[192 x 64] = w^T[192x49] * corr^T[49x64] + b  via
//            V_WMMA_F32_16X16X4_F32 (K padded 49 -> 52 = 13 wmma steps).
template <int C, int H, int W>
__global__ __launch_bounds__(128) void cost_tok_kernel(
    const float* __restrict__ lvl,   // [T,B,C,H,W]
    const float* __restrict__ wmat,  // [49,192]
    const float* __restrict__ bias,  // [192]
    float* __restrict__ out,         // [T-1,B,192,H,W]
    int B) {
  constexpr int HW  = H * W;
  constexpr int KP  = 52;   // padded correlation-offset count (49 -> 52)
  constexpr int TD  = 192;  // token dim
  constexpr int PIX = 64;   // pixels per block

  __shared__ float f1_lds[C * PIX];     // [c][p]
  __shared__ float w_lds[KP * TD];      // [k][d], rows 49..51 zero
  __shared__ float b_lds[TD];
  __shared__ float corr_lds[PIX * KP];  // [p][k]

  const int tid   = threadIdx.x;
  const int tb    = blockIdx.y;          // t*B + b, t in [0, T-2]
  const int pbase = blockIdx.x * PIX;    // first flattened pixel of tile

  const float* f1 = lvl + (size_t)tb * C * HW;
  const float* f2 = lvl + (size_t)(tb + B) * C * HW;  // frame t+1, same b

  // ---- stage weights (zero-padded), bias, f1 tile into LDS ----
  for (int i = tid; i < KP * TD; i += 128)
    w_lds[i] = (i < 49 * TD) ? wmat[i] : 0.0f;
  for (int i = tid; i < TD; i += 128)
    b_lds[i] = bias[i];
  for (int i = tid; i < C * PIX; i += 128) {
    int c = i >> 6, p = i & 63;
    f1_lds[i] = f1[(size_t)c * HW + pbase + p];
  }
  __syncthreads();

  // ---- phase 1: correlation dots ----
  {
    const float scale = 1.0f / __builtin_sqrtf((float)C);
    const int p  = tid & 63;       // pixel within tile
    const int kl = tid >> 6;       // 0 or 1: two k's per pass
    const int pg = pbase + p;
    const int h  = pg / W;
    const int w  = pg % W;
    const float* f1p = f1_lds + p;

    for (int kbase = 0; kbase < KP; kbase += 2) {
      const int k = kbase + kl;
      float acc = 0.0f;
      if (k < 49) {
        const int dy = k / 7 - 3;
        const int dx = k % 7 - 3;
        int nh = h - dy; nh += (nh < 0) ? H : 0; nh -= (nh >= H) ? H : 0;
        int nw = w - dx; nw += (nw < 0) ? W : 0; nw -= (nw >= W) ? W : 0;
        const float* f2p = f2 + (size_t)nh * W + nw;
#pragma unroll 4
        for (int c = 0; c < C; ++c)
          acc += f1p[c * PIX] * f2p[(size_t)c * HW];
        acc *= scale;
      }
      corr_lds[p * KP + k] = acc;   // pad columns written as 0
    }
  }
  __syncthreads();

  // ---- phase 2: WMMA f32 GEMM  tok^T = w^T * corr^T + b ----
  {
    const int lane = tid & 31;
    const int wave = tid >> 5;
    const int p0   = wave * 16;     // this wave's 16-pixel N-tile
    const int nlo  = lane & 15;
    const int lhi  = lane >> 4;     // 0: K={4k,4k+1} half, 1: K={4k+2,4k+3}

    // Pre-load B fragments (corr^T, K x 16 pixels) for all 13 K-steps.
    // 32-bit B 4x16 layout: v0 = K0 (lanes 0-15) / K2 (lanes 16-31), v1 = K1/K3.
    v2f bfr[13];
#pragma unroll
    for (int ks = 0; ks < 13; ++ks) {
      const int krow = 4 * ks + 2 * lhi;
      const float* src = &corr_lds[(p0 + nlo) * KP + krow];
      bfr[ks].x = src[0];
      bfr[ks].y = src[1];
    }

    for (int m = 0; m < 12; ++m) {
      const int d0 = m * 16;        // this M-tile's token-dim base
      // Accumulator = bias: D[M=d][N=pixel]; vgpr r holds d = d0 + r + 8*lhi.
      v8f acc;
#pragma unroll
      for (int r = 0; r < 8; ++r)
        acc[r] = b_lds[d0 + r + 8 * lhi];

#pragma unroll
      for (int ks = 0; ks < 13; ++ks) {
        const int krow = 4 * ks + 2 * lhi;
        // A fragment (w^T, 16 d x 4 K): lane holds rows d0+nlo, K pair per lhi.
        v2f afr;
        afr.x = w_lds[krow * TD + d0 + nlo];
        afr.y = w_lds[(krow + 1) * TD + d0 + nlo];
        acc = __builtin_amdgcn_wmma_f32_16x16x4_f32(
            false, afr, false, bfr[ks], (short)0, acc, false, false);
      }

      // Store: contiguous 16 pixels per half-wave per vgpr row -> coalesced.
      float* op = out + ((size_t)tb * TD + d0) * HW + pbase + p0 + nlo;
#pragma unroll
      for (int r = 0; r < 8; ++r)
        op[(size_t)(r + 8 * lhi) * HW] = acc[r];
    }
  }
}

extern "C" void kernel_launch(void* const* d_in, const int* in_sizes, int n_in,
                              void* d_out, int out_size, void* d_ws, size_t ws_size,
                              hipStream_t stream) {
  const float* l1 = (const float*)d_in[0];
  const float* l2 = (const float*)d_in[1];
  const float* l3 = (const float*)d_in[2];
  const float* w1 = (const float*)d_in[3];
  const float* b1 = (const float*)d_in[4];
  const float* w2 = (const float*)d_in[5];
  const float* b2 = (const float*)d_in[6];
  const float* w3 = (const float*)d_in[7];
  const float* b3 = (const float*)d_in[8];

  const int T = 3, B = 2, TD = 192;
  float* tok1 = (float*)d_out;
  float* tok2 = tok1 + (size_t)(T - 1) * B * TD * 128 * 128;
  float* tok3 = tok2 + (size_t)(T - 1) * B * TD * 64 * 64;

  dim3 blk(128);
  cost_tok_kernel<64, 128, 128>
      <<<dim3(128 * 128 / 64, (T - 1) * B), blk, 0, stream>>>(l1, w1, b1, tok1, B);
  cost_tok_kernel<128, 64, 64>
      <<<dim3(64 * 64 / 64, (T - 1) * B), blk, 0, stream>>>(l2, w2, b2, tok2, B);
  cost_tok_kernel<192, 32, 32>
      <<<dim3(32 * 32 / 64, (T - 1) * B), blk, 0, stream>>>(l3, w3, b3, tok3, B);
}